// GCN_46188078301659
// MI455X (gfx1250) — compile-verified
//
#include <hip/hip_runtime.h>
#include <hip/hip_bf16.h>

#define NN 50000
#define NE 640000
#define D  128
#define NL 5
#define NG 512

typedef __attribute__((ext_vector_type(2))) float v2f;
typedef __attribute__((ext_vector_type(8))) float v8f;

// ---------------------------------------------------------------------------
// GEMM: Out[rows x 128] = X[rows x 128] @ W[128 x 128] (+ bias), f32 WMMA.
// Block = 256 threads (8 waves), computes a 128-row stripe. Wave w owns rows
// [w*16, w*16+16) x all 128 cols via 8 accumulator tiles (16x16 each).
// K consumed in 4 LDS-staged panels of 32; each panel = 8 WMMA k-steps (K=4).
// W panel stored K-pair-interleaved so each B fragment is one aligned
// ds_load_b64; X panel stride 34 keeps A-frag half-groups on disjoint banks.
// ---------------------------------------------------------------------------
__global__ __launch_bounds__(256)
void gemm128_wmma(const float* __restrict__ X, const float* __restrict__ W,
                  const float* __restrict__ bias, float* __restrict__ Out,
                  int rows)
{
    __shared__ float sX[128][34];       // [row][k]     (A fragments)
    __shared__ float sWp[16][144][2];   // [kpair][n][j] = W[k0+2*kpair+j][n]

    const int tid  = threadIdx.x;
    const int wave = tid >> 5;
    const int lane = tid & 31;
    const int half = lane >> 4;
    const int l16  = lane & 15;
    const int blockRow = blockIdx.x * 128;

    v8f acc[8] = {};

    for (int k0 = 0; k0 < 128; k0 += 32) {
        // Stage X panel [128 rows x 32 k] (coalesced float4, zero-pad OOB rows)
        {
            int r = tid >> 3;          // 0..31
            int c = (tid & 7) * 4;     // 0..28
            #pragma unroll
            for (int it = 0; it < 4; ++it) {
                int row = r + it * 32;
                int gr  = blockRow + row;
                float4 v = make_float4(0.f, 0.f, 0.f, 0.f);
                if (gr < rows) v = *(const float4*)(X + (size_t)gr * D + k0 + c);
                sX[row][c+0] = v.x; sX[row][c+1] = v.y;
                sX[row][c+2] = v.z; sX[row][c+3] = v.w;
            }
        }
        // Stage W panel [32 k x 128 n], pair-interleaved over K
        {
            int k = tid >> 5;          // 0..7
            int c = (tid & 31) * 4;    // 0..124
            #pragma unroll
            for (int it = 0; it < 4; ++it) {
                int kk = k + it * 8;   // 0..31
                float4 v = *(const float4*)(W + (size_t)(k0 + kk) * D + c);
                int kp = kk >> 1, j = kk & 1;
                sWp[kp][c+0][j] = v.x; sWp[kp][c+1][j] = v.y;
                sWp[kp][c+2][j] = v.z; sWp[kp][c+3][j] = v.w;
            }
        }
        __syncthreads();

        #pragma unroll
        for (int kk = 0; kk < 8; ++kk) {
            // this half-wave's K pair: k = kk*4 + half*2 + {0,1}
            const int ka = kk * 4 + half * 2;
            const int kp = ka >> 1;                    // = kk*2 + half
            v2f a = *(const v2f*)&sX[wave * 16 + l16][ka];
            #pragma unroll
            for (int nt = 0; nt < 8; ++nt) {
                v2f b = *(const v2f*)&sWp[kp][nt * 16 + l16][0];
                acc[nt] = __builtin_amdgcn_wmma_f32_16x16x4_f32(
                    false, a, false, b, (short)0, acc[nt], false, false);
            }
        }
        __syncthreads();
    }

    // Store: acc[nt] VGPR r holds C[m = r + 8*half][n = nt*16 + l16]
    #pragma unroll
    for (int nt = 0; nt < 8; ++nt) {
        int col = nt * 16 + l16;
        float bv = bias ? bias[col] : 0.0f;
        #pragma unroll
        for (int r = 0; r < 8; ++r) {
            int grow = blockRow + wave * 16 + r + 8 * half;
            if (grow < rows) Out[(size_t)grow * D + col] = acc[nt][r] + bv;
        }
    }
}

// ---------------------------------------------------------------------------
// CSR build + helpers
// ---------------------------------------------------------------------------
__global__ void hist_kernel(const int* __restrict__ dst, int* __restrict__ cnt)
{
    int e = blockIdx.x * blockDim.x + threadIdx.x;
    if (e < NE) atomicAdd(&cnt[dst[e]], 1);
}

__global__ __launch_bounds__(1024)
void scan_kernel(const int* __restrict__ cnt, int* __restrict__ rowptr,
                 float* __restrict__ deg, int* __restrict__ fill)
{
    __shared__ int buf[1024];
    __shared__ int s_total;
    const int tid = threadIdx.x;
    if (tid == 0) s_total = 0;
    __syncthreads();
    for (int base = 0; base < NN; base += 1024) {
        int idx = base + tid;
        int v = (idx < NN) ? cnt[idx] : 0;
        buf[tid] = v;
        __syncthreads();
        #pragma unroll
        for (int off = 1; off < 1024; off <<= 1) {
            int t = (tid >= off) ? buf[tid - off] : 0;
            __syncthreads();
            buf[tid] += t;
            __syncthreads();
        }
        int incl = buf[tid];
        int tot  = s_total;
        if (idx < NN) {
            rowptr[idx] = tot + incl - v;
            deg[idx]    = (float)(v > 0 ? v : 1);
            fill[idx]   = 0;
        }
        __syncthreads();
        if (tid == 1023) s_total = tot + buf[1023];
        __syncthreads();
    }
    if (tid == 0) rowptr[NN] = s_total;  // == NE
}

__global__ void csr_fill_kernel(const int* __restrict__ src, const int* __restrict__ dst,
                                const int* __restrict__ rowptr, int* __restrict__ fill,
                                int* __restrict__ csr_src, int* __restrict__ csr_eid)
{
    int e = blockIdx.x * blockDim.x + threadIdx.x;
    if (e >= NE) return;
    int d = dst[e];
    int pos = atomicAdd(&fill[d], 1);
    int idx = rowptr[d] + pos;
    csr_src[idx] = src[e];
    csr_eid[idx] = e;
}

// EA[i] = sum over in-edges of edge_attr[e]  (one wave per node, float4 lanes,
// dual accumulators for memory-level parallelism)
__global__ __launch_bounds__(256)
void ea_kernel(const float* __restrict__ edge_attr, const int* __restrict__ rowptr,
               const int* __restrict__ csr_eid, float* __restrict__ EA)
{
    int node = blockIdx.x * 8 + (threadIdx.x >> 5);
    if (node >= NN) return;
    int lane = threadIdx.x & 31;
    int beg = rowptr[node], end = rowptr[node + 1];
    float4 s0 = make_float4(0.f, 0.f, 0.f, 0.f);
    float4 s1 = make_float4(0.f, 0.f, 0.f, 0.f);
    int e = beg;
    for (; e + 1 < end; e += 2) {
        int e0 = csr_eid[e], e1 = csr_eid[e + 1];
        float4 m0 = ((const float4*)(edge_attr + (size_t)e0 * D))[lane];
        float4 m1 = ((const float4*)(edge_attr + (size_t)e1 * D))[lane];
        s0.x += m0.x; s0.y += m0.y; s0.z += m0.z; s0.w += m0.w;
        s1.x += m1.x; s1.y += m1.y; s1.z += m1.z; s1.w += m1.w;
    }
    if (e < end) {
        int e0 = csr_eid[e];
        float4 m0 = ((const float4*)(edge_attr + (size_t)e0 * D))[lane];
        s0.x += m0.x; s0.y += m0.y; s0.z += m0.z; s0.w += m0.w;
    }
    s0.x += s1.x; s0.y += s1.y; s0.z += s1.z; s0.w += s1.w;
    ((float4*)(EA + (size_t)node * D))[lane] = s0;
}

__global__ void init_h_kernel(const float* __restrict__ node_emb, float* __restrict__ h)
{
    int i = blockIdx.x * blockDim.x + threadIdx.x;
    if (i < NN * D) h[i] = node_emb[i & (D - 1)];
}

// h[i] = (base[i] + sum_{e in CSR row} hw[src_e]) / deg[i] + hw[i]  (+ReLU)
__global__ __launch_bounds__(256)
void agg_kernel(const float* __restrict__ hw, const float* __restrict__ base,
                const int* __restrict__ rowptr, const int* __restrict__ csr_src,
                const float* __restrict__ deg, float* __restrict__ h, int relu)
{
    int node = blockIdx.x * 8 + (threadIdx.x >> 5);
    if (node >= NN) return;
    int lane = threadIdx.x & 31;
    int beg = rowptr[node], end = rowptr[node + 1];
    float4 s0 = ((const float4*)(base + (size_t)node * D))[lane];
    float4 s1 = make_float4(0.f, 0.f, 0.f, 0.f);
    int e = beg;
    for (; e + 1 < end; e += 2) {
        int n0 = csr_src[e], n1 = csr_src[e + 1];
        float4 m0 = ((const float4*)(hw + (size_t)n0 * D))[lane];
        float4 m1 = ((const float4*)(hw + (size_t)n1 * D))[lane];
        s0.x += m0.x; s0.y += m0.y; s0.z += m0.z; s0.w += m0.w;
        s1.x += m1.x; s1.y += m1.y; s1.z += m1.z; s1.w += m1.w;
    }
    if (e < end) {
        int n0 = csr_src[e];
        float4 m0 = ((const float4*)(hw + (size_t)n0 * D))[lane];
        s0.x += m0.x; s0.y += m0.y; s0.z += m0.z; s0.w += m0.w;
    }
    s0.x += s1.x; s0.y += s1.y; s0.z += s1.z; s0.w += s1.w;

    float dg = deg[node];
    float4 hv = ((const float4*)(hw + (size_t)node * D))[lane];
    float4 o;
    o.x = s0.x / dg + hv.x; o.y = s0.y / dg + hv.y;
    o.z = s0.z / dg + hv.z; o.w = s0.w / dg + hv.w;
    if (relu) {
        o.x = fmaxf(o.x, 0.f); o.y = fmaxf(o.y, 0.f);
        o.z = fmaxf(o.z, 0.f); o.w = fmaxf(o.w, 0.f);
    }
    ((float4*)(h + (size_t)node * D))[lane] = o;
}

__global__ __launch_bounds__(256)
void pool_kernel(const float* __restrict__ h, const int* __restrict__ batch,
                 float* __restrict__ pooled, int* __restrict__ gcnt)
{
    int node = blockIdx.x * 8 + (threadIdx.x >> 5);
    if (node >= NN) return;
    int lane = threadIdx.x & 31;
    int g = batch[node];
    float4 v = ((const float4*)(h + (size_t)node * D))[lane];
    float* p = pooled + (size_t)g * D + lane * 4;
    atomicAdd(p + 0, v.x); atomicAdd(p + 1, v.y);
    atomicAdd(p + 2, v.z); atomicAdd(p + 3, v.w);
    if (lane == 0) atomicAdd(&gcnt[g], 1);
}

__global__ void pmean_kernel(const float* __restrict__ pooled, const int* __restrict__ gcnt,
                             float* __restrict__ pm)
{
    int i = blockIdx.x * blockDim.x + threadIdx.x;
    if (i >= NG * D) return;
    int g = i / D;
    float c = (float)gcnt[g];
    pm[i] = pooled[i] / fmaxf(c, 1.0f);
}

// ---------------------------------------------------------------------------
extern "C" void kernel_launch(void* const* d_in, const int* in_sizes, int n_in,
                              void* d_out, int out_size, void* d_ws, size_t ws_size,
                              hipStream_t stream)
{
    const int*   edge_index = (const int*)d_in[1];
    const int*   srcp = edge_index;
    const int*   dstp = edge_index + NE;
    const float* edge_attr = (const float*)d_in[2];
    const int*   batch = (const int*)d_in[3];
    const float* node_emb = (const float*)d_in[4];
    const float* Ws = (const float*)d_in[5];
    const float* bs = (const float*)d_in[6];
    const float* Wes = (const float*)d_in[7];
    const float* Wp = (const float*)d_in[8];
    const float* bp = (const float*)d_in[9];
    float* out = (float*)d_out;

    char* ws = (char*)d_ws;
    size_t off = 0;
    auto ALLOC = [&](size_t bytes) -> void* {
        void* p = ws + off;
        off += (bytes + 255) & ~(size_t)255;
        return p;
    };
    float* h      = (float*)ALLOC((size_t)NN * D * 4);
    float* hw     = (float*)ALLOC((size_t)NN * D * 4);
    float* base   = (float*)ALLOC((size_t)NN * D * 4);
    float* EA     = (float*)ALLOC((size_t)NN * D * 4);
    float* pooled = (float*)ALLOC((size_t)NG * D * 4);
    float* pm     = (float*)ALLOC((size_t)NG * D * 4);
    float* deg    = (float*)ALLOC((size_t)NN * 4);
    int*   gcnt   = (int*)  ALLOC((size_t)NG * 4);
    int*   cnt    = (int*)  ALLOC((size_t)NN * 4);
    int*   fill   = (int*)  ALLOC((size_t)NN * 4);
    int*   rowptr = (int*)  ALLOC((size_t)(NN + 1) * 4);
    int*   csr_s  = (int*)  ALLOC((size_t)NE * 4);
    int*   csr_e  = (int*)  ALLOC((size_t)NE * 4);

    // --- one-time graph preprocessing ---
    hipMemsetAsync(cnt, 0, (size_t)NN * 4, stream);
    hist_kernel<<<(NE + 255) / 256, 256, 0, stream>>>(dstp, cnt);
    scan_kernel<<<1, 1024, 0, stream>>>(cnt, rowptr, deg, fill);
    csr_fill_kernel<<<(NE + 255) / 256, 256, 0, stream>>>(srcp, dstp, rowptr, fill, csr_s, csr_e);
    ea_kernel<<<(NN + 7) / 8, 256, 0, stream>>>(edge_attr, rowptr, csr_e, EA);
    init_h_kernel<<<(NN * D + 255) / 256, 256, 0, stream>>>(node_emb, h);

    // --- layers ---
    const int gblocks = (NN + 127) / 128;
    for (int l = 0; l < NL; ++l) {
        gemm128_wmma<<<gblocks, 256, 0, stream>>>(h,  Ws  + (size_t)l * D * D, bs + (size_t)l * D, hw,   NN);
        gemm128_wmma<<<gblocks, 256, 0, stream>>>(EA, Wes + (size_t)l * D * D, nullptr,            base, NN);
        agg_kernel<<<(NN + 7) / 8, 256, 0, stream>>>(hw, base, rowptr, csr_s, deg, h, (l < NL - 1) ? 1 : 0);
    }

    // --- pooling + projection ---
    hipMemsetAsync(pooled, 0, (size_t)NG * D * 4, stream);
    hipMemsetAsync(gcnt, 0, (size_t)NG * 4, stream);
    pool_kernel<<<(NN + 7) / 8, 256, 0, stream>>>(h, batch, pooled, gcnt);
    pmean_kernel<<<(NG * D + 255) / 256, 256, 0, stream>>>(pooled, gcnt, pm);
    gemm128_wmma<<<(NG + 127) / 128, 256, 0, stream>>>(pm, Wp, bp, out, NG);
}